// SCTLossEuclidean_42992622633552
// MI455X (gfx1250) — compile-verified
//
#include <hip/hip_runtime.h>
#include <hip/hip_bf16.h>

// SCT loss, MI455X (gfx1250), wave32.
//
// The reference's "hardest" mining provably degenerates: the -1 masked fills
// are always the row minima (all real distances >= 1e-6), so argmin picks the
// first masked column. Hence every row only needs its distance to 4 fixed
// reference rows (rows 0, 1, 4096, 4097 of the concatenated input). We compute
// those dot products on the f32 WMMA path (16x16x4), K=128 accumulated over
// 32 WMMA steps per 16-row tile, fully bandwidth-bound (4 MB total traffic).

typedef __attribute__((ext_vector_type(2))) float v2f;
typedef __attribute__((ext_vector_type(8))) float v8f;

namespace {
constexpr int kP = 4096;   // positives
constexpr int kN = 8192;   // total rows
constexpr int kD = 128;    // feature dim
}

__global__ __launch_bounds__(256)
void sct_dots_kernel(const float* __restrict__ pos,
                     const float* __restrict__ neg,
                     float* __restrict__ term /* [kN] */)
{
    const int tid   = threadIdx.x;
    const int wave  = tid >> 5;          // 0..7
    const int lane  = tid & 31;
    const int nhalf = lane & 15;         // A: row-in-tile, B: column id
    const int khi   = lane >> 4;         // 0 -> K {0,1}; 1 -> K {2,3}
    const int koff  = khi * 2;

    const int blockRow0 = blockIdx.x * 128;        // 64 blocks of 128 rows
    const int rowBase   = blockRow0 + wave * 16;   // 16 rows per wave

    // Whole 128-row block lies entirely inside pos or neg (4096 % 128 == 0).
    const float* base = (blockRow0 < kP) ? (pos + (size_t)blockRow0 * kD)
                                         : (neg + (size_t)(blockRow0 - kP) * kD);
    const float* myRow = base + (size_t)(wave * 16 + nhalf) * kD;

    // Reference rows 0,1,4096,4097 mapped to B columns 0..3; columns 4..15 are
    // zeroed via bmask (lanes still load a safe address -> branchless, EXEC
    // stays all-ones around the WMMA).
    const float* refRow = (nhalf == 1) ? (pos + kD)
                        : (nhalf == 2) ? neg
                        : (nhalf == 3) ? (neg + kD)
                                       : pos;
    const float bmask = (nhalf < 4) ? 1.0f : 0.0f;

    v8f  acc = {};     // 16x16 f32 accumulator (8 VGPRs)
    float sA = 0.0f;   // this lane's K-half of the A-row self-dot
    float sB = 0.0f;   // this lane's K-half of the ref-column self-dot

    #pragma unroll 4
    for (int k = 0; k < kD; k += 4) {
        v2f a, b;
        a.x = myRow[k + koff];
        a.y = myRow[k + koff + 1];
        b.x = refRow[k + koff]     * bmask;
        b.y = refRow[k + koff + 1] * bmask;
        sA = fmaf(a.x, a.x, fmaf(a.y, a.y, sA));
        sB = fmaf(b.x, b.x, fmaf(b.y, b.y, sB));
        // D = A(16x4) * B(4x16) + C ; f32 WMMA, 8-arg form:
        // (neg_a, A, neg_b, B, c_mod, C, reuse_a, reuse_b)
        acc = __builtin_amdgcn_wmma_f32_16x16x4_f32(
            false, a, false, b, (short)0, acc, false, false);
    }

    // Combine the two K-halves (lane <-> lane^16) of the self-dots.
    sA += __shfl_xor(sA, 16, 32);
    sB += __shfl_xor(sB, 16, 32);

    // C/D layout: VGPR r -> (M=r, N=lane) on lanes 0-15, (M=8+r, N=lane-16)
    // on lanes 16-31. Unpack through LDS for per-row scalar math.
    __shared__ float lds_dot[8][16][16];   // [wave][M][Ncol]
    __shared__ float lds_sA[8][16];        // [wave][M]
    __shared__ float lds_sB[8][4];         // [wave][refcol]

    #pragma unroll
    for (int r = 0; r < 8; ++r)
        lds_dot[wave][khi * 8 + r][nhalf] = acc[r];
    if (khi == 0)               lds_sA[wave][nhalf] = sA;
    if (khi == 0 && nhalf < 4)  lds_sB[wave][nhalf] = sB;
    __syncthreads();

    if (khi == 0) {
        const int M  = nhalf;
        const int gi = rowBase + M;                 // global row index
        const float inv_ni = rsqrtf(lds_sA[wave][M]);

        // Degenerate mining indices (first masked column):
        //   Pos: first different-label col; Neg: first same-label off-diag col.
        int pref, nref;
        if (gi < kP) { pref = 2; nref = (gi == 0)  ? 1 : 0; }
        else         { pref = 0; nref = (gi == kP) ? 3 : 2; }

        const float cosp = lds_dot[wave][M][pref] * inv_ni * rsqrtf(lds_sB[wave][pref]);
        const float cosn = lds_dot[wave][M][nref] * inv_ni * rsqrtf(lds_sB[wave][nref]);

        // dist = sqrt(max(||a||^2 + ||b||^2 - 2 a.b, 1e-12)), unit rows -> 2 - 2cos
        const float Pos = sqrtf(fmaxf(2.0f - 2.0f * cosp, 1e-12f));
        const float Neg = sqrtf(fmaxf(2.0f - 2.0f * cosn, 1e-12f));

        // -log_softmax0([Pos,Neg]/T) = softplus((Neg-Pos)/T), T = 0.1
        const float x  = (Neg - Pos) * 10.0f;
        const float sp = (x > 0.0f) ? (x + log1pf(__expf(-x))) : log1pf(__expf(x));
        term[gi] = sp;
    }
}

__global__ __launch_bounds__(256)
void sct_reduce_kernel(const float* __restrict__ term, float* __restrict__ out)
{
    __shared__ float red[256];
    float s = 0.0f;
    for (int i = threadIdx.x; i < kN; i += 256) s += term[i];   // fixed order
    red[threadIdx.x] = s;
    __syncthreads();
    for (int off = 128; off > 0; off >>= 1) {
        if (threadIdx.x < off) red[threadIdx.x] += red[threadIdx.x + off];
        __syncthreads();
    }
    // Mask is provably all-valid -> denominator is kN.
    if (threadIdx.x == 0) out[0] = red[0] / (float)kN;
}

extern "C" void kernel_launch(void* const* d_in, const int* in_sizes, int n_in,
                              void* d_out, int out_size, void* d_ws, size_t ws_size,
                              hipStream_t stream)
{
    const float* pos = (const float*)d_in[0];   // [4096,128] f32
    const float* neg = (const float*)d_in[1];   // [4096,128] f32
    float* term = (float*)d_ws;                 // 8192 f32 scratch (32 KB)

    sct_dots_kernel<<<64, 256, 0, stream>>>(pos, neg, term);
    sct_reduce_kernel<<<1, 256, 0, stream>>>(term, (float*)d_out);
}